// GaussianBlur3D_47296179863889
// MI455X (gfx1250) — compile-verified
//
#include <hip/hip_runtime.h>

typedef float v2f __attribute__((ext_vector_type(2)));
typedef float v8f __attribute__((ext_vector_type(8)));

// Tensor: (B=2, C=4, D=128, H=128, W=128) f32, NCDHW flat.
// total = 16,777,216 elements. planes (s,d) = 1024, each 128x128.

#define TOTAL   16777216
#define NPLANE  1024        // (B*C) * D
#define PLANESZ 16384       // 128*128

// ---------------------------------------------------------------------------
// Pass along a strided axis (D: shift=14, H: shift=7).
// One thread per output element; w (contiguous) is the fast thread dim, so
// all 7 taps are fully coalesced. Intermediates stay in the 192MB L2.
// ---------------------------------------------------------------------------
__global__ __launch_bounds__(256) void gauss_pass_strided(
    const float* __restrict__ src, float* __restrict__ dst,
    const float* __restrict__ g, int shift)
{
    int tid = blockIdx.x * 256 + threadIdx.x;
    int c = (tid >> shift) & 127;
    int stride = 1 << shift;
    int base = tid - (c << shift);        // flat index with axis coord zeroed
    float acc = 0.0f;
#pragma unroll
    for (int k = 0; k < 7; ++k) {
        int cc = c + k - 3;
        cc = cc < 0 ? 0 : (cc > 127 ? 127 : cc);   // replication pad
        acc = fmaf(g[k], src[base + cc * stride], acc);
    }
    dst[tid] = acc;
}

// Branch-free band select: g[j] for j in [0,6], else 0 (v_cndmask chain).
__device__ __forceinline__ float gsel(int j, float g0, float g1, float g2,
                                      float g3, float g4, float g5, float g6)
{
    float r = 0.0f;
    r = (j == 0) ? g0 : r;
    r = (j == 1) ? g1 : r;
    r = (j == 2) ? g2 : r;
    r = (j == 3) ? g3 : r;
    r = (j == 4) ? g4 : r;
    r = (j == 5) ? g5 : r;
    r = (j == 6) ? g6 : r;
    return r;
}

__device__ __forceinline__ v8f wmma_acc(v2f a, v2f b, v8f c)
{
    // (neg_a, A, neg_b, B, c_mod, C, reuse_a, reuse_b)
    return __builtin_amdgcn_wmma_f32_16x16x4_f32(
        false, a, false, b, (short)0, c, false, false);
}

// ---------------------------------------------------------------------------
// Pass along W (contiguous axis) as a banded matmul on the WMMA pipe:
//   out_tile(16x16) = In_ext(16x24) x G(24x16),  G[p][w] = g[p-w] (band 0..6)
// One wave per 16(h) x 128(w) row strip = 8 tiles; the constant B fragments
// are built once per wave (cndmask chains from scalar-loaded g) and reused
// across all 8 tiles. tx is wave-uniform, so edge-pad clamping sits behind a
// uniform branch; the 6 interior tiles are clamp-free b64 loads + 6 WMMAs.
// ---------------------------------------------------------------------------
__global__ __launch_bounds__(256) void gauss_pass_w_wmma(
    const float* __restrict__ src, float* __restrict__ dst,
    const float* __restrict__ gptr)
{
    const int lane  = threadIdx.x & 31;
    const int wave  = blockIdx.x * (256 / 32) + (threadIdx.x >> 5);
    const int ty    = wave & 7;           // 8 row strips per plane
    const int plane = wave >> 3;          // 1024 planes
    const int hbase = ty * 16;

    const int M    = lane & 15;           // A/D row (h), B/D col (w)
    const int half = lane >> 4;

    // g is uniform: literal-index loads -> scalar loads, broadcast.
    const float g0 = gptr[0], g1 = gptr[1], g2 = gptr[2], g3 = gptr[3];
    const float g4 = gptr[4], g5 = gptr[5], g6 = gptr[6];

    // Constant banded-matrix fragments, built once per wave.
    v2f bfrag[6];
#pragma unroll
    for (int c = 0; c < 6; ++c) {
        const int k0 = 4 * c + 2 * half;  // this lane's K pair: k0, k0+1
        const int j0 = k0 - M;            // N == lane&15
        bfrag[c].x = gsel(j0,     g0, g1, g2, g3, g4, g5, g6);
        bfrag[c].y = gsel(j0 + 1, g0, g1, g2, g3, g4, g5, g6);
    }

    const float* row   = src + plane * PLANESZ + (hbase + M) * 128;
    float*       qcol  = dst + plane * PLANESZ + M;   // column N = lane&15

    for (int tx = 0; tx < 8; ++tx) {
        const int wbase = tx * 16;
        v8f acc = {0.f, 0.f, 0.f, 0.f, 0.f, 0.f, 0.f, 0.f};

        if (tx == 0 || tx == 7) {
            // Edge tiles: replication-pad clamp per element.
#pragma unroll
            for (int c = 0; c < 6; ++c) {
                const int k0 = 4 * c + 2 * half;
                int w0 = wbase + k0 - 3;
                int w1 = w0 + 1;
                int cw0 = w0 < 0 ? 0 : (w0 > 127 ? 127 : w0);
                int cw1 = w1 < 0 ? 0 : (w1 > 127 ? 127 : w1);
                v2f a;
                a.x = row[cw0];
                a.y = row[cw1];
                acc = wmma_acc(a, bfrag[c], acc);
            }
        } else {
            // Interior tiles: clamp-free; a.x/a.y merge into global_load_b64.
            const float* rp = row + wbase - 3;
#pragma unroll
            for (int c = 0; c < 6; ++c) {
                const int k0 = 4 * c + 2 * half;
                v2f a;
                a.x = rp[k0];
                a.y = rp[k0 + 1];
                acc = wmma_acc(a, bfrag[c], acc);
            }
        }

        // D layout: VGPR r -> row (r + 8*half), col N (coalesced per VGPR).
#pragma unroll
        for (int r = 0; r < 8; ++r)
            qcol[(hbase + r + 8 * half) * 128 + wbase] = acc[r];
    }
}

// ---------------------------------------------------------------------------
// d_in[0] = x (16.7M f32), d_in[1] = g (7 f32), d_in[2] = sigma (unused).
// Pipeline: blurD: x -> out ; blurH: out -> ws ; blurW(WMMA): ws -> out.
// Needs only 64MB of d_ws; intermediates stay resident in the 192MB L2.
// ---------------------------------------------------------------------------
extern "C" void kernel_launch(void* const* d_in, const int* in_sizes, int n_in,
                              void* d_out, int out_size, void* d_ws, size_t ws_size,
                              hipStream_t stream) {
    const float* x = (const float*)d_in[0];
    const float* g = (const float*)d_in[1];
    float* out = (float*)d_out;
    float* tmp = (float*)d_ws;

    const int blocks = TOTAL / 256;                 // 65536

    // Pass 1: blur along D (stride 2^14)
    gauss_pass_strided<<<blocks, 256, 0, stream>>>(x, out, g, 14);
    // Pass 2: blur along H (stride 2^7)
    gauss_pass_strided<<<blocks, 256, 0, stream>>>(out, tmp, g, 7);
    // Pass 3: blur along W via f32 WMMA banded matmul, 8 tiles per wave
    const int wmma_blocks = (NPLANE * 8) / 8;       // 1024 blocks x 8 waves
    gauss_pass_w_wmma<<<wmma_blocks, 256, 0, stream>>>(tmp, out, g);
}